// CausalConvWithHiddenState_5136780886277
// MI455X (gfx1250) — compile-verified
//
#include <hip/hip_runtime.h>
#include <cstdint>

#ifndef __has_builtin
#define __has_builtin(x) 0
#endif

// Problem constants (from reference setup_inputs)
constexpr int Bc   = 4;
constexpr int Dc   = 4096;
constexpr int Lc   = 4096;
constexpr int Kc   = 3;            // d_conv - 1 (state length)
constexpr int LOUT = Lc + Kc;      // 4099 outputs per row
constexpr int TPB  = 256;          // 8 wave32 waves
constexpr int F4_PER_THREAD = (Lc / 4) / TPB;  // 4 float4 chunks per thread

// gcc-style vector type matching the builtin's expected parameter type
typedef int v4i __attribute__((vector_size(16)));
typedef __attribute__((address_space(1))) v4i* global_v4i_ptr;
typedef __attribute__((address_space(3))) v4i* lds_v4i_ptr;

__global__ __launch_bounds__(TPB)
void causal_conv_rows(const float* __restrict__ x,
                      const float* __restrict__ w,
                      const float* __restrict__ bias,
                      const float* __restrict__ state,
                      float* __restrict__ out)
{
    // [0] pad, [1..3] state, [4..4099] x (16B aligned), [4100..4102] zero tail
    __shared__ float lds_f[Lc + 8];

    const int r   = blockIdx.x;        // r = b*D + d
    const int d   = r & (Dc - 1);
    const int tid = threadIdx.x;

    const float* xrow = x + (size_t)r * Lc;

    // ---- Stage the whole row into LDS via CDNA5 async global->LDS DMA ----
    // Each lane moves four 16B chunks; lds dest 16 + 16*k is 16B aligned.
#pragma unroll
    for (int c = 0; c < F4_PER_THREAD; ++c) {
        const int k = c * TPB + tid;   // float4 index within the row
#if __has_builtin(__builtin_amdgcn_global_load_async_to_lds_b128)
        __builtin_amdgcn_global_load_async_to_lds_b128(
            (global_v4i_ptr)(xrow + (size_t)k * 4),
            (lds_v4i_ptr)(&lds_f[4 + k * 4]),
            /*offset=*/0, /*cpol=*/0);
#else
        unsigned ldsaddr =
            (unsigned)(uintptr_t)(__attribute__((address_space(3))) void*)(&lds_f[4 + k * 4]);
        asm volatile("global_load_async_to_lds_b128 %0, %1, off"
                     :: "v"(ldsaddr), "v"(xrow + (size_t)k * 4)
                     : "memory");
#endif
    }

    // Learnable-state prefix + zero tail (plain DS stores; barrier orders them)
    if (tid < Kc) {
        lds_f[1 + tid]      = state[d * Kc + tid];
        lds_f[4 + Lc + tid] = 0.0f;
    }

#if __has_builtin(__builtin_amdgcn_s_wait_asynccnt)
    __builtin_amdgcn_s_wait_asynccnt(0);
#else
    asm volatile("s_wait_asynccnt 0x0" ::: "memory");
#endif
    __syncthreads();

    // Per-row filter + bias: uniform per block -> scalar loads
    const float w0 = w[d * 4 + 0];
    const float w1 = w[d * 4 + 1];
    const float w2 = w[d * 4 + 2];
    const float w3 = w[d * 4 + 3];
    const float bs = bias[d];

    float* orow = out + (size_t)r * LOUT;

    // y[t] = w0*xp[t] + w1*xp[t+1] + w2*xp[t+2] + w3*xp[t+3] + bias,
    // xp[j] == lds_f[1+j]. Interleaved mapping: lane-consecutive LDS reads
    // (bank-conflict-free) and coalesced NT stores.
#pragma unroll
    for (int c = 0; c < Lc / TPB; ++c) {
        const int t = c * TPB + tid;
        const float* p = &lds_f[1 + t];
        float acc = fmaf(w0, p[0], bs);
        acc = fmaf(w1, p[1], acc);
        acc = fmaf(w2, p[2], acc);
        acc = fmaf(w3, p[3], acc);
        __builtin_nontemporal_store(acc, orow + t);
    }
    if (tid < Kc) {                     // remainder outputs t = L..L+2
        const int t = Lc + tid;
        const float* p = &lds_f[1 + t];
        float acc = fmaf(w0, p[0], bs);
        acc = fmaf(w1, p[1], acc);
        acc = fmaf(w2, p[2], acc);
        acc = fmaf(w3, p[3], acc);
        __builtin_nontemporal_store(acc, orow + t);
    }
}

extern "C" void kernel_launch(void* const* d_in, const int* in_sizes, int n_in,
                              void* d_out, int out_size, void* d_ws, size_t ws_size,
                              hipStream_t stream)
{
    const float* x     = (const float*)d_in[0];   // [B, D, L]
    const float* w     = (const float*)d_in[1];   // [D, 1, 4]
    const float* bias  = (const float*)d_in[2];   // [D]
    const float* state = (const float*)d_in[3];   // [D, 3]
    float* out = (float*)d_out;                   // [B, D, L+3]
    (void)in_sizes; (void)n_in; (void)out_size; (void)d_ws; (void)ws_size;

    causal_conv_rows<<<dim3(Bc * Dc), dim3(TPB), 0, stream>>>(x, w, bias, state, out);
}